// TripletLoss_50474455662593
// MI455X (gfx1250) — compile-verified
//
#include <hip/hip_runtime.h>

#define MARGIN 0.3f

typedef float v2f __attribute__((ext_vector_type(2)));
typedef float v8f __attribute__((ext_vector_type(8)));

// Extract replicated-per-half diagonal-ish row constant from a C fragment:
// register r holds row r (lanes 0-15, col=lane) and row r+8 (lanes 16-31,
// col=lane-16). Diagonal of row q lives at lane q (q<8) / lane q+16 (q>=8),
// i.e. source lane r + 24*h for register r, half h.
__device__ __forceinline__ float diag_bcast(float c, int r, int h) {
  return __shfl(c, r + 24 * h, 32);
}

// One wave (32 lanes) per 16-row block of `a`.
// Everything is WMMA: dist_ap = diag(A_blk @ P_blk^T), fallback = diag of the
// block-diagonal tile of A @ N^T, then scan 16-column tiles of an = a @ n^T in
// order, mining the first column below threshold per row, early-exiting when
// all 16 rows have found a semi-hard negative.
__global__ __launch_bounds__(32) void triplet_semihard_kernel(
    const float* __restrict__ A, const float* __restrict__ P,
    const float* __restrict__ N, float* __restrict__ row_loss, int B) {
  constexpr int D = 128;
  const int lane = threadIdx.x;   // 0..31 (wave32)
  const int m    = lane & 15;     // row/col within tile
  const int h    = lane >> 4;     // half of the wave
  const int i0   = blockIdx.x * 16;

  // ---- A fragments for v_wmma_f32_16x16x4_f32 over K=128 (32 chunks) ----
  // 16x4 f32 A layout: lane m holds row m; VGPR v holds K = kbase + 2*h + v
  const float* abase = A + (size_t)(i0 + m) * D;
  v2f afrag[32];
#pragma unroll
  for (int c = 0; c < 32; ++c)
    afrag[c] = *(const v2f*)(abase + 4 * c + 2 * h);

  // ---- dist_ap = diag(A_blk @ P_blk^T) via WMMA ----
  const float* pbase = P + (size_t)(i0 + m) * D;
  v2f bfrag[32];
#pragma unroll
  for (int c = 0; c < 32; ++c)
    bfrag[c] = *(const v2f*)(pbase + 4 * c + 2 * h);
  v8f accP = {0.f, 0.f, 0.f, 0.f, 0.f, 0.f, 0.f, 0.f};
#pragma unroll
  for (int c = 0; c < 32; ++c)
    accP = __builtin_amdgcn_wmma_f32_16x16x4_f32(
        false, afrag[c], false, bfrag[c], (short)0, accP, false, false);

  // ---- fallback an[i,i] = diag of block-diagonal tile of A @ N^T ----
  const float* ndbase = N + (size_t)(i0 + m) * D;
#pragma unroll
  for (int c = 0; c < 32; ++c)
    bfrag[c] = *(const v2f*)(ndbase + 4 * c + 2 * h);
  v8f accD = {0.f, 0.f, 0.f, 0.f, 0.f, 0.f, 0.f, 0.f};
#pragma unroll
  for (int c = 0; c < 32; ++c)
    accD = __builtin_amdgcn_wmma_f32_16x16x4_f32(
        false, afrag[c], false, bfrag[c], (short)0, accD, false, false);

  // ---- per-register row constants (row = r + 8*h, replicated per half) ----
  float dap[8], thr[8], anv[8];
#pragma unroll
  for (int r = 0; r < 8; ++r) {
    dap[r] = diag_bcast(accP[r], r, h);
    thr[r] = dap[r] + MARGIN;
    anv[r] = diag_bcast(accD[r], r, h);   // preload diagonal fallback
  }
  unsigned foundmask = 0;                 // bit r: row r+8*h has a negative

  const int ntiles = B >> 4;
  for (int jt = 0; jt < ntiles; ++jt) {
    // B = n^T tile: B[k][j] = n[j0+j][k] -> same per-lane pattern as A
    const float* nbase = N + (size_t)(jt * 16 + m) * D;
    // prefetch next tile's row for this lane while we compute
    __builtin_prefetch(nbase + 16 * D, 0, 1);

    // load the whole tile first (32 independent b64 loads -> one clause),
    // then run 32 back-to-back WMMAs
#pragma unroll
    for (int c = 0; c < 32; ++c)
      bfrag[c] = *(const v2f*)(nbase + 4 * c + 2 * h);
    v8f acc = {0.f, 0.f, 0.f, 0.f, 0.f, 0.f, 0.f, 0.f};
#pragma unroll
    for (int c = 0; c < 32; ++c)
      acc = __builtin_amdgcn_wmma_f32_16x16x4_f32(
          false, afrag[c], false, bfrag[c], (short)0, acc, false, false);

    // ---- mining: first column (bit order == column order) below threshold ----
#pragma unroll
    for (int r = 0; r < 8; ++r) {
      const bool cond    = acc[r] < thr[r];
      const unsigned msk = __builtin_amdgcn_ballot_w32(cond);
      const unsigned hm  = h ? (msk >> 16) : (msk & 0xFFFFu);
      const int col = hm ? (int)__builtin_ctz(hm) : 0;
      const float v = __shfl(acc[r], col + 16 * h, 32);  // an[row][first col]
      if (hm && !((foundmask >> r) & 1u)) {
        foundmask |= 1u << r;
        anv[r] = v;
      }
    }
    // wave-uniform early exit (keeps EXEC all-ones for WMMA)
    if (__builtin_amdgcn_ballot_w32(foundmask == 0xFFu) == 0xFFFFFFFFu) break;
  }

  // each half's lane 0 writes its 8 rows (state replicated across half)
  if (m == 0) {
#pragma unroll
    for (int r = 0; r < 8; ++r) {
      const float l = anv[r] - dap[r] + MARGIN;
      row_loss[i0 + 8 * h + r] = l > 0.f ? l : 0.f;
    }
  }
}

__global__ __launch_bounds__(256) void reduce_mean_kernel(
    const float* __restrict__ row_loss, float* __restrict__ out, int B) {
  __shared__ float sm[256];
  float s = 0.f;
  for (int i = threadIdx.x; i < B; i += 256) s += row_loss[i];
  sm[threadIdx.x] = s;
  __syncthreads();
#pragma unroll
  for (int off = 128; off > 0; off >>= 1) {
    if ((int)threadIdx.x < off) sm[threadIdx.x] += sm[threadIdx.x + off];
    __syncthreads();
  }
  if (threadIdx.x == 0) out[0] = sm[0] / (float)B;
}

extern "C" void kernel_launch(void* const* d_in, const int* in_sizes, int n_in,
                              void* d_out, int out_size, void* d_ws, size_t ws_size,
                              hipStream_t stream) {
  const float* a = (const float*)d_in[0];
  const float* p = (const float*)d_in[1];
  const float* n = (const float*)d_in[2];
  const int B = in_sizes[0] / 128;        // 8192
  float* row_loss = (float*)d_ws;         // B floats of scratch

  triplet_semihard_kernel<<<dim3(B / 16), dim3(32), 0, stream>>>(
      a, p, n, row_loss, B);
  reduce_mean_kernel<<<dim3(1), dim3(256), 0, stream>>>(
      row_loss, (float*)d_out, B);
}